// LocalAffinity_15350213115958
// MI455X (gfx1250) — compile-verified
//
#include <hip/hip_runtime.h>

typedef float f4 __attribute__((ext_vector_type(4)));
typedef __attribute__((address_space(1))) int* gptr_t;
typedef __attribute__((address_space(3))) int* lptr_t;

namespace {
constexpr int KCH  = 2;
constexpr int DIMZ = 64, DIMY = 128, DIMX = 128;
constexpr int TZ = 4, TY = 8, TX = 64;            // per-block tile
constexpr int HALO = 2;                           // dilation-2 reach
constexpr int LZ = TZ + 2 * HALO;                 // 8
constexpr int LYN = TY + 2 * HALO;                // 12
constexpr int LXN = 72;                           // padded row: x-lds = gx - x0 + 4 (16B-aligned centers)
constexpr int XOFF = 4;
constexpr int LDS_N = LZ * LYN * LXN;             // 6912 floats = 27 KB (of 320 KB/WGP)
constexpr int NTAP = 52;                          // 26 offsets x 2 dilations
constexpr long long VOL = (long long)DIMZ * DIMY * DIMX;  // 1048576 per (k,c) slab
}

#define ASYNC_LDS __has_builtin(__builtin_amdgcn_global_load_async_to_lds_b32)

__global__ __launch_bounds__(256) void LocalAffinity52_kernel(const float* __restrict__ xin,
                                                              float* __restrict__ out) {
  __shared__ float tile[LDS_N];

  int id = blockIdx.x;
  const int bx = id & 1;   id >>= 1;   // DIMX/TX = 2
  const int by = id & 15;  id >>= 4;   // DIMY/TY = 16
  const int bz = id & 15;  id >>= 4;   // DIMZ/TZ = 16
  const int k  = id;                   // KCH = 2
  const int x0 = bx * TX, y0 = by * TY, z0 = bz * TZ;
  const int tid = threadIdx.x;
  const float* __restrict__ xk = xin + (long long)k * VOL;

  // ---- stage haloed tile into LDS (replicate padding == clamped global index) ----
  // 6912 elements / 256 threads = 27 full iterations, consecutive lanes -> consecutive gx (coalesced).
#pragma unroll
  for (int it = 0; it < LDS_N / 256; ++it) {
    const int t  = tid + it * 256;
    const int lx = t % LXN;
    const int r  = t / LXN;
    const int ly = r % LYN;
    const int lz = r / LYN;
    const int gz = min(max(z0 + lz - HALO, 0), DIMZ - 1);
    const int gy = min(max(y0 + ly - HALO, 0), DIMY - 1);
    const int gx = min(max(x0 + lx - XOFF, 0), DIMX - 1);
    const float* gp = xk + ((long long)gz * DIMY + gy) * DIMX + gx;
#if ASYNC_LDS
    // gfx1250 async DMA: global -> LDS, tracked by ASYNCcnt (no VGPR data staging)
    __builtin_amdgcn_global_load_async_to_lds_b32(
        (gptr_t)(gp), (lptr_t)(&tile[t]), 0, 0);
#else
    tile[t] = *gp;
#endif
  }
#if ASYNC_LDS
#if __has_builtin(__builtin_amdgcn_s_wait_asynccnt)
  __builtin_amdgcn_s_wait_asynccnt(0);
#else
  asm volatile("s_wait_asynccnt 0" ::: "memory");
#endif
#endif
  __syncthreads();

  // ---- compute 52 taps: out = center - neighbor, 128b NT stores (output >> L2, keep L2 for input) ----
  // Tile = 2048 elems = 512 float4; 256 threads -> 2 rounds. Lanes 0..15 cover one 64-float row.
#pragma unroll
  for (int itc = 0; itc < (TZ * TY * TX / 4) / 256; ++itc) {
    const int idx = tid + itc * 256;
    const int tx4 = idx & 15;          // float4 slot in x (16 per 64-wide row)
    const int ty  = (idx >> 4) & 7;
    const int tz  = idx >> 7;
    const int lxb = tx4 * 4 + XOFF;    // 16B-aligned LDS column
    const int lyc = ty + HALO;
    const int lzc = tz + HALO;
    const f4 ctr = *(const f4*)&tile[(lzc * LYN + lyc) * LXN + lxb];

    const int gz = z0 + tz, gy = y0 + ty, gxb = x0 + tx4 * 4;
    float* op = out + (long long)(k * NTAP) * VOL + ((long long)gz * DIMY + gy) * DIMX + gxb;

    int c = 0;
#pragma unroll
    for (int d = 1; d <= 2; ++d) {
#pragma unroll
      for (int oi = 0; oi < 27; ++oi) {
        if (oi == 13) continue;        // skip center tap
        const int dz = oi / 9 - 1;
        const int dy = (oi / 3) % 3 - 1;
        const int dx = oi % 3 - 1;
        const int nb = ((lzc + dz * d) * LYN + (lyc + dy * d)) * LXN + (lxb + dx * d);
        f4 n;
        if (dx == 0) {                 // constant-folded per unrolled tap: aligned b128 LDS read
          n = *(const f4*)&tile[nb];
        } else {
          n = (f4){tile[nb + 0], tile[nb + 1], tile[nb + 2], tile[nb + 3]};
        }
        const f4 r = ctr - n;
        __builtin_nontemporal_store(r, (f4*)(op + (long long)c * VOL));
        ++c;
      }
    }
  }
}

extern "C" void kernel_launch(void* const* d_in, const int* in_sizes, int n_in,
                              void* d_out, int out_size, void* d_ws, size_t ws_size,
                              hipStream_t stream) {
  (void)in_sizes; (void)n_in; (void)d_ws; (void)ws_size; (void)out_size;
  const float* x = (const float*)d_in[0];
  float* out = (float*)d_out;
  const int nblocks = (DIMX / TX) * (DIMY / TY) * (DIMZ / TZ) * KCH;  // 2*16*16*2 = 1024
  LocalAffinity52_kernel<<<nblocks, 256, 0, stream>>>(x, out);
}